// TripletMiningLoss_45045617000831
// MI455X (gfx1250) — compile-verified
//
#include <hip/hip_runtime.h>
#include <hip/hip_bf16.h>
#include <stdint.h>
#include <math.h>

#define B_N    8192
#define D_DIM  128
#define MARGIN 0.3f
#define EPS_C  1e-6f
#define TINY_F 1.17549435e-38f

typedef __attribute__((ext_vector_type(2))) float v2f;
typedef __attribute__((ext_vector_type(8))) float v8f;

// ---------------- JAX threefry-2x32 (exact) ----------------
__host__ __device__ inline void tf_round(uint32_t& x0, uint32_t& x1, int r) {
    x0 += x1;
    x1 = (x1 << r) | (x1 >> (32 - r));
    x1 ^= x0;
}

__host__ __device__ inline void threefry2x32(uint32_t k0, uint32_t k1,
                                             uint32_t c0, uint32_t c1,
                                             uint32_t& o0, uint32_t& o1) {
    uint32_t ks2 = k0 ^ k1 ^ 0x1BD11BDAu;
    uint32_t x0 = c0 + k0, x1 = c1 + k1;
    tf_round(x0, x1, 13); tf_round(x0, x1, 15); tf_round(x0, x1, 26); tf_round(x0, x1, 6);
    x0 += k1;  x1 += ks2 + 1u;
    tf_round(x0, x1, 17); tf_round(x0, x1, 29); tf_round(x0, x1, 16); tf_round(x0, x1, 24);
    x0 += ks2; x1 += k0 + 2u;
    tf_round(x0, x1, 13); tf_round(x0, x1, 15); tf_round(x0, x1, 26); tf_round(x0, x1, 6);
    x0 += k0;  x1 += k1 + 3u;
    tf_round(x0, x1, 17); tf_round(x0, x1, 29); tf_round(x0, x1, 16); tf_round(x0, x1, 24);
    x0 += k1;  x1 += ks2 + 4u;
    tf_round(x0, x1, 13); tf_round(x0, x1, 15); tf_round(x0, x1, 26); tf_round(x0, x1, 6);
    x0 += ks2; x1 += k0 + 5u;
    o0 = x0; o1 = x1;
}

// gumbel[flat] for a (8192,8192) draw: JAX pairs count `flat` with `flat +/- half`.
__device__ inline float jax_gumbel(uint32_t k0, uint32_t k1, uint32_t flat) {
    const uint32_t HALF = 1u << 25;  // 8192*8192/2
    uint32_t o0, o1, bits;
    if (flat < HALF) { threefry2x32(k0, k1, flat, flat + HALF, o0, o1); bits = o0; }
    else             { threefry2x32(k0, k1, flat - HALF, flat, o0, o1); bits = o1; }
    float f = __uint_as_float((bits >> 9) | 0x3f800000u) - 1.0f;  // [0,1)
    float u = fmaxf(f, TINY_F);                                    // uniform(minval=tiny)
    return -logf(-logf(u));
}

// ---------------- main fused kernel: 1 workgroup = 16 anchors ----------------
// fp32 WMMA 16x16x4 operand layouts:
//   A 16x4 : lane(0-15)=M, v0:K=2*half, v1:K=2*half+1  -> float2 per K-step of 4
//   B 4x16 : lane&15=N,    v0:K=2*half, v1:K=2*half+1  (rows r, r+2 per VGPR)
//   C 16x16: vr: lanes0-15 M=r, lanes16-31 M=r+8, N=lane&15
__global__ __launch_bounds__(256)
void triplet_main(const float* __restrict__ emb, const int* __restrict__ labels,
                  const float* __restrict__ sq, float* __restrict__ accum,
                  uint32_t kp0, uint32_t kp1, uint32_t kn0, uint32_t kn1) {
    __shared__ float s_tile[8][16][16];
    __shared__ float s_f0[256];
    __shared__ int   s_i0[256];
    __shared__ float s_f1[256];
    __shared__ float s_f2[256];
    __shared__ int   s_i1[256];
    __shared__ float s_ap[16];
    __shared__ int   s_pidx[16], s_pok[16], s_hni[16], s_nok[16], s_nidx[16];
    __shared__ float s_sqrow[16];
    __shared__ int   s_lab[16];

    const int tid  = threadIdx.x;
    const int w    = tid >> 5;
    const int l    = tid & 31;
    const int n16  = l & 15;
    const int half = l >> 4;
    const int r0   = blockIdx.x * 16;

    if (tid < 16) { s_sqrow[tid] = sq[r0 + tid]; s_lab[tid] = labels[r0 + tid]; }
    __syncthreads();

    // Preload A fragments (row block), reused by both passes.
    v2f afrag[32];
    const float* arow = emb + (size_t)(r0 + n16) * D_DIM + 2 * half;
#pragma unroll
    for (int kk = 0; kk < 32; ++kk) afrag[kk] = *(const v2f*)(arow + kk * 4);

    const int i_row = r0 + n16;   // anchor row mined by this lane
    const int li    = s_lab[n16];
    const int colb  = 8 * half;   // this lane scans 8 cols of the 16-wide tile

    // Double-buffered B fragments (ping-pong kept register-resident by 2x unroll).
    v2f b0[32], b1[32];

    auto loadB = [&](v2f (&buf)[32], int c0) {
        const float* brow = emb + (size_t)(c0 + n16) * D_DIM + 2 * half;
#pragma unroll
        for (int kk = 0; kk < 32; ++kk) buf[kk] = *(const v2f*)(brow + kk * 4);
    };

    // GEMM a 16x16 tile, convert to distances, park in this wave's LDS slab.
    auto gemm = [&](const v2f (&buf)[32], int c0) {
        v8f acc0 = {0.f, 0.f, 0.f, 0.f, 0.f, 0.f, 0.f, 0.f};
        v8f acc1 = {0.f, 0.f, 0.f, 0.f, 0.f, 0.f, 0.f, 0.f};
#pragma unroll
        for (int kk = 0; kk < 32; kk += 2) {
            acc0 = __builtin_amdgcn_wmma_f32_16x16x4_f32(false, afrag[kk], false, buf[kk],
                                                         (short)0, acc0, false, false);
            acc1 = __builtin_amdgcn_wmma_f32_16x16x4_f32(false, afrag[kk + 1], false, buf[kk + 1],
                                                         (short)0, acc1, false, false);
        }
        float sqj = sq[c0 + n16];
#pragma unroll
        for (int r = 0; r < 8; ++r) {
            int rm = r + 8 * half;
            float d2 = s_sqrow[rm] + sqj - 2.0f * (acc0[r] + acc1[r]);
            s_tile[w][rm][n16] = sqrtf(fmaxf(d2, 0.0f));
        }
        asm volatile("s_wait_dscnt 0" ::: "memory");  // cross-lane LDS RAW inside wave
    };

    // ---------- pass 1: random positive (Gumbel argmax) + hardest negative ----------
    float gpg = -__builtin_inff(); int gpi = 0x7FFFFFFF; float gpd = 0.0f;
    float ngd =  __builtin_inff(); int ngi = 0x7FFFFFFF;

    auto mine1 = [&](int c0) {
#pragma unroll 1
        for (int cc = 0; cc < 8; ++cc) {
            int   j  = c0 + colb + cc;
            float dv = s_tile[w][n16][colb + cc];
            int   lj = labels[j];
            if (lj == li) {
                if (j != i_row) {
                    float g = jax_gumbel(kp0, kp1, (uint32_t)i_row * (uint32_t)B_N + (uint32_t)j);
                    if (g > gpg) { gpg = g; gpi = j; gpd = dv; }
                }
            } else {
                if (dv < ngd) { ngd = dv; ngi = j; }
            }
        }
    };

    // tile t = tt*8 + w  ->  c0 = tt*128 + w*16
    loadB(b0, w * 16);
    for (int tt = 0; tt < 64; tt += 2) {
        const int c0a = tt * 128 + w * 16;
        const int c0b = c0a + 128;
        loadB(b1, c0b);                       // prefetch overlaps gemm+mine of tile a
        gemm(b0, c0a);
        mine1(c0a);
        if (tt + 2 < 64) loadB(b0, c0a + 256);  // prefetch overlaps gemm+mine of tile b
        gemm(b1, c0b);
        mine1(c0b);
    }

    s_f0[tid] = gpg; s_i0[tid] = gpi; s_f1[tid] = gpd; s_f2[tid] = ngd; s_i1[tid] = ngi;
    __syncthreads();
    if (tid < 16) {  // combine 8 waves x 2 half-lanes per row; first-index tie-break
        float bg = -__builtin_inff(); int bi = 0x7FFFFFFF; float bd = 0.0f;
        float nd =  __builtin_inff(); int ni = 0x7FFFFFFF;
        for (int s = 0; s < 16; ++s) {
            int src = (s >> 1) * 32 + (s & 1) * 16 + tid;
            float cg = s_f0[src]; int ci = s_i0[src];
            if (cg > bg || (cg == bg && ci < bi)) { bg = cg; bi = ci; bd = s_f1[src]; }
            float cd = s_f2[src]; int cni = s_i1[src];
            if (cd < nd || (cd == nd && cni < ni)) { nd = cd; ni = cni; }
        }
        s_ap[tid]   = bd;
        s_pidx[tid] = bi;
        s_pok[tid]  = (bg > -__builtin_inff()) ? 1 : 0;
        s_hni[tid]  = ni;
        s_nok[tid]  = (nd < __builtin_inff()) ? 1 : 0;
    }
    __syncthreads();

    // ---------- pass 2: semi-hard negatives (ap < d < ap + margin), Gumbel argmax ----------
    float gng = -__builtin_inff(); int gni = 0x7FFFFFFF;
    const float apr = s_ap[n16];

    auto mine2 = [&](int c0) {
#pragma unroll 1
        for (int cc = 0; cc < 8; ++cc) {
            int   j  = c0 + colb + cc;
            float dv = s_tile[w][n16][colb + cc];
            int   lj = labels[j];
            if (lj != li && dv > apr && dv < apr + MARGIN) {
                float g = jax_gumbel(kn0, kn1, (uint32_t)i_row * (uint32_t)B_N + (uint32_t)j);
                if (g > gng) { gng = g; gni = j; }
            }
        }
    };

    loadB(b0, w * 16);
    for (int tt = 0; tt < 64; tt += 2) {
        const int c0a = tt * 128 + w * 16;
        const int c0b = c0a + 128;
        loadB(b1, c0b);
        gemm(b0, c0a);
        mine2(c0a);
        if (tt + 2 < 64) loadB(b0, c0a + 256);
        gemm(b1, c0b);
        mine2(c0b);
    }

    s_f0[tid] = gng; s_i0[tid] = gni;
    __syncthreads();
    if (tid < 16) {
        float bg = -__builtin_inff(); int bi = 0x7FFFFFFF;
        for (int s = 0; s < 16; ++s) {
            int src = (s >> 1) * 32 + (s & 1) * 16 + tid;
            float cg = s_f0[src]; int ci = s_i0[src];
            if (cg > bg || (cg == bg && ci < bi)) { bg = cg; bi = ci; }
        }
        s_nidx[tid] = (bg > -__builtin_inff()) ? bi : s_hni[tid];
    }
    __syncthreads();

    // ---------- final per-anchor triplet loss (wave 0) ----------
    if (w == 0) {
        float lsum = 0.0f, lcnt = 0.0f;
        for (int r = 0; r < 16; ++r) {
            if (!(s_pok[r] && s_nok[r])) continue;  // 'valid' mask
            const float* A_ = emb + (size_t)(r0 + r) * D_DIM;
            const float* P_ = emb + (size_t)s_pidx[r] * D_DIM;
            const float* N_ = emb + (size_t)s_nidx[r] * D_DIM;
            float sp = 0.f, sn = 0.f;
#pragma unroll
            for (int q = 0; q < 4; ++q) {
                int   k  = l * 4 + q;
                float av = A_[k];
                float dp = av - P_[k] + EPS_C; sp += dp * dp;
                float dn = av - N_[k] + EPS_C; sn += dn * dn;
            }
#pragma unroll
            for (int off = 16; off > 0; off >>= 1) {
                sp += __shfl_down(sp, off);
                sn += __shfl_down(sn, off);
            }
            if (l == 0) {
                float per = sqrtf(sp) - sqrtf(sn) + MARGIN;
                lsum += fmaxf(per, 0.0f);
                lcnt += 1.0f;
            }
        }
        if (l == 0 && lcnt > 0.0f) {
            atomicAdd(&accum[0], lsum);
            atomicAdd(&accum[1], lcnt);
        }
    }
}

// ---------------- init: zero accumulators, per-row squared norms ----------------
__global__ void triplet_init(const float* __restrict__ emb, float* __restrict__ ws) {
    int tid = blockIdx.x * blockDim.x + threadIdx.x;
    if (tid == 0) { ws[0] = 0.0f; ws[1] = 0.0f; }
    if (tid < B_N) {
        const float* row = emb + (size_t)tid * D_DIM;
        float s = 0.0f;
        for (int k = 0; k < D_DIM; k += 4) {
            float4 v = *(const float4*)(row + k);
            s += v.x * v.x + v.y * v.y + v.z * v.z + v.w * v.w;
        }
        ws[2 + tid] = s;
    }
}

__global__ void triplet_final(const float* __restrict__ ws, float* __restrict__ out) {
    if (blockIdx.x == 0 && threadIdx.x == 0)
        out[0] = ws[0] / fmaxf(ws[1], 1.0f);
}

extern "C" void kernel_launch(void* const* d_in, const int* in_sizes, int n_in,
                              void* d_out, int out_size, void* d_ws, size_t ws_size,
                              hipStream_t stream) {
    const float* emb    = (const float*)d_in[0];
    const int*   labels = (const int*)d_in[1];
    float*       ws     = (float*)d_ws;
    float*       out    = (float*)d_out;

    // Host-side JAX PRNG: key(42) = {0,42}; split -> counts iota(4) paired as (0,2),(1,3)
    uint32_t a0, a1, b0, b1;
    threefry2x32(0u, 42u, 0u, 2u, a0, a1);
    threefry2x32(0u, 42u, 1u, 3u, b0, b1);
    const uint32_t kp0 = a0, kp1 = b0;   // first row of reshape(2,2)
    const uint32_t kn0 = a1, kn1 = b1;   // second row

    triplet_init<<<(B_N + 255) / 256, 256, 0, stream>>>(emb, ws);
    triplet_main<<<B_N / 16, 256, 0, stream>>>(emb, labels, ws + 2, ws,
                                               kp0, kp1, kn0, kn1);
    triplet_final<<<1, 32, 0, stream>>>(ws, out);
}